// Net_51196010168472
// MI455X (gfx1250) — compile-verified
//
#include <hip/hip_runtime.h>
#include <hip/hip_bf16.h>
#include <math.h>

typedef float v2f __attribute__((ext_vector_type(2)));
typedef float v8f __attribute__((ext_vector_type(8)));

#define N_NODES  80000
#define N_EDGES  1280000
#define N_GRAPHS 512
#define F        64

// ---------------------------------------------------------------- utilities
__global__ void zero_f32(float* __restrict__ p, int n) {
    int i = blockIdx.x * blockDim.x + threadIdx.x;
    if (i < n) p[i] = 0.0f;
}

__global__ void deg_kernel(const int* __restrict__ dst, float* __restrict__ deg) {
    int e = blockIdx.x * blockDim.x + threadIdx.x;
    if (e < N_EDGES) atomicAdd(&deg[dst[e]], 1.0f);
}

// =====================================================================
// Layer-1 node matmuls (Fin = 3, padded to one K=4 WMMA step).
// Branch-free: load all valid elements, lane-select into WMMA layout.
// f32 16x16x4 operand layout (wave32):
//   A (16x4): lane m=L%16; the 2 VGPRs hold K = (L/16)*2, (L/16)*2+1
//   B (4x16): lane n=L%16; the 2 VGPRs hold K rows (L/16)*2, (L/16)*2+1
//   C (16x16): VGPR r: lanes 0-15 -> M=r, lanes 16-31 -> M=r+8; N=lane%16
// =====================================================================
__global__ void __launch_bounds__(128)
node_mm3(const float* __restrict__ x,
         const float* __restrict__ W0, const float* __restrict__ W1,
         const float* __restrict__ Wr, const float* __restrict__ bias,
         float* __restrict__ y0, float* __restrict__ y1, float* __restrict__ yr) {
    const int wave  = threadIdx.x >> 5;
    const int lane  = threadIdx.x & 31;
    const int mtile = blockIdx.x * 4 + wave;          // 80000/16 = 5000 tiles
    const int m16   = lane & 15;
    const int half  = lane >> 4;                       // 0: K=0,1  1: K=2,(3=pad)
    const int row   = mtile * 16 + m16;

    const float h0 = x[(size_t)row * 3 + 0];
    const float h1 = x[(size_t)row * 3 + 1];
    const float h2 = x[(size_t)row * 3 + 2];
    v2f a;
    a.x = half ? h2   : h0;
    a.y = half ? 0.0f : h1;

    v8f acc[3][4];
    #pragma unroll
    for (int mt = 0; mt < 3; ++mt)
        #pragma unroll
        for (int nt = 0; nt < 4; ++nt)
            #pragma unroll
            for (int r = 0; r < 8; ++r) acc[mt][nt][r] = 0.0f;

    const float* Ws[3] = { W0, W1, Wr };
    #pragma unroll
    for (int mt = 0; mt < 3; ++mt) {
        const float* __restrict__ W = Ws[mt];
        #pragma unroll
        for (int nt = 0; nt < 4; ++nt) {
            const int n = nt * 16 + m16;
            const float w0r = W[n];            // K=0 row
            const float w1r = W[64 + n];       // K=1 row
            const float w2r = W[128 + n];      // K=2 row
            v2f b;
            b.x = half ? w2r  : w0r;
            b.y = half ? 0.0f : w1r;
            acc[mt][nt] = __builtin_amdgcn_wmma_f32_16x16x4_f32(
                false, a, false, b, (short)0, acc[mt][nt], false, false);
        }
    }

    #pragma unroll
    for (int nt = 0; nt < 4; ++nt) {
        const int n = nt * 16 + m16;
        const float bn = bias[n];
        #pragma unroll
        for (int r = 0; r < 8; ++r) {
            const int m    = r + half * 8;
            const size_t o = (size_t)(mtile * 16 + m) * F + n;
            y0[o] = acc[0][nt][r];
            y1[o] = acc[1][nt][r];
            yr[o] = acc[2][nt][r] + bn;
        }
    }
}

// =====================================================================
// Hidden-layer node matmuls (Fin = 64). The three 64x64 weight matrices
// (48KB) have rows 0-15 (12KB) staged into LDS once per workgroup with the CDNA5 async
// global->LDS mover (GLOBAL_LOAD_ASYNC_TO_LDS_B128, ASYNCcnt), then each
// of the 4 waves runs 16 K-steps of V_WMMA_F32_16X16X4_F32 with B
// operands served from LDS.
// =====================================================================
__global__ void __launch_bounds__(128)
node_mm64(const float* __restrict__ hin,
          const float* __restrict__ W0, const float* __restrict__ W1,
          const float* __restrict__ Wr, const float* __restrict__ bias,
          float* __restrict__ y0, float* __restrict__ y1, float* __restrict__ yr) {
    __shared__ float ldsW[3 * 64 * 64];    // row-major mirror of W0 | W1 | Wr

    // ---- async stage: 3072 floats = 768 x 16B chunks; 128 threads x 6 ----
    const float* Wg[3] = { W0, W1, Wr };
    #pragma unroll
    for (int mat = 0; mat < 3; ++mat) {
        #pragma unroll
        for (int c = 0; c < 2; ++c) {
            const int chunk = threadIdx.x + c * 128;          // 0..255
            const float* g = Wg[mat] + chunk * 4;              // 16B aligned
            // generic LDS pointer keeps the LDS byte offset in addr[31:0]
            const unsigned ldsAddr =
                (unsigned)(unsigned long long)(const void*)&ldsW[mat * 4096 + chunk * 4];
            asm volatile("global_load_async_to_lds_b128 %0, %1, off"
                         :: "v"(ldsAddr), "v"(g)
                         : "memory");
        }
    }
    asm volatile("s_wait_asynccnt 0x0" ::: "memory");  // this wave's copies done
    __syncthreads();                                    // all waves' copies done

    const int wave  = threadIdx.x >> 5;
    const int lane  = threadIdx.x & 31;
    const int mtile = blockIdx.x * 4 + wave;
    const int m16   = lane & 15;
    const int half  = lane >> 4;
    const int row   = mtile * 16 + m16;

    const float2* __restrict__ arow = (const float2*)(hin + (size_t)row * F);

    v8f acc[3][4];
    #pragma unroll
    for (int mt = 0; mt < 3; ++mt)
        #pragma unroll
        for (int nt = 0; nt < 4; ++nt)
            #pragma unroll
            for (int r = 0; r < 8; ++r) acc[mt][nt][r] = 0.0f;

    #pragma unroll
    for (int ks = 0; ks < 16; ++ks) {
        const int k0 = ks * 4 + half * 2;             // K rows this lane feeds
        const float2 av = arow[ks * 2 + half];        // A pair (K=k0, k0+1)
        v2f a; a.x = av.x; a.y = av.y;
        #pragma unroll
        for (int mt = 0; mt < 3; ++mt) {
            #pragma unroll
            for (int nt = 0; nt < 4; ++nt) {
                const int n = nt * 16 + m16;
                v2f b;
                b.x = ldsW[mt * 4096 + k0 * 64 + n];
                b.y = ldsW[mt * 4096 + (k0 + 1) * 64 + n];
                acc[mt][nt] = __builtin_amdgcn_wmma_f32_16x16x4_f32(
                    false, a, false, b, (short)0, acc[mt][nt], false, false);
            }
        }
    }

    #pragma unroll
    for (int nt = 0; nt < 4; ++nt) {
        const int n = nt * 16 + m16;
        const float bn = bias[n];
        #pragma unroll
        for (int r = 0; r < 8; ++r) {
            const int m    = r + half * 8;
            const size_t o = (size_t)(mtile * 16 + m) * F + n;
            y0[o] = acc[0][nt][r];
            y1[o] = acc[1][nt][r];
            yr[o] = acc[2][nt][r] + bn;
        }
    }
}

// ------------------------------------------------------------- edge scatter
// one wave per edge; each lane handles 2 of 64 feature columns (float2)
__global__ void __launch_bounds__(256)
edge_kernel(const int* __restrict__ src, const int* __restrict__ dst,
            const float* __restrict__ pseudo,
            const float* __restrict__ y0, const float* __restrict__ y1,
            float* __restrict__ agg) {
    const int e = blockIdx.x * (blockDim.x >> 5) + (threadIdx.x >> 5);
    if (e >= N_EDGES) return;
    const int lane = threadIdx.x & 31;
    const int s = src[e];
    const int d = dst[e];
    const float u  = pseudo[e];
    const float w0 = 1.0f - u;

    const float2 a0 = ((const float2*)(y0 + (size_t)s * F))[lane];
    const float2 a1 = ((const float2*)(y1 + (size_t)s * F))[lane];
    float* out = agg + (size_t)d * F + lane * 2;
    atomicAdd(out,     a0.x * w0 + a1.x * u);
    atomicAdd(out + 1, a0.y * w0 + a1.y * u);
}

// --------------------------------------------------- mean-norm + root + ELU
// Also clears agg in place so the next layer starts from zero.
__global__ void combine_kernel(float* __restrict__ agg,
                               const float* __restrict__ yr,
                               const float* __restrict__ deg,
                               float* __restrict__ hout) {
    const int idx = blockIdx.x * blockDim.x + threadIdx.x;
    if (idx >= N_NODES * F) return;
    const int i = idx >> 6;
    const float dg = fmaxf(deg[i], 1.0f);
    const float v  = agg[idx] / dg + yr[idx];
    agg[idx]  = 0.0f;                           // ready for next layer
    hout[idx] = (v > 0.0f) ? v : expm1f(v);     // ELU(alpha=1)
}

// ------------------------------------------------------------- graph pooling
__global__ void pool_kernel(const float* __restrict__ h,
                            const int* __restrict__ batch,
                            float* __restrict__ pooled, float* __restrict__ cnt) {
    const int idx = blockIdx.x * blockDim.x + threadIdx.x;
    if (idx >= N_NODES * F) return;
    const int i = idx >> 6, c = idx & 63;
    const int g = batch[i];
    atomicAdd(&pooled[(size_t)g * F + c], h[idx]);
    if (c == 0) atomicAdd(&cnt[g], 1.0f);
}

// ------------------------------------------------------ FC head + logsoftmax
__global__ void head_kernel(const float* __restrict__ pooled,
                            const float* __restrict__ cnt,
                            const float* __restrict__ fc_w,
                            const float* __restrict__ fc_b,
                            float* __restrict__ out) {
    const int g = blockIdx.x * blockDim.x + threadIdx.x;
    if (g >= N_GRAPHS) return;
    const float inv = 1.0f / fmaxf(cnt[g], 1.0f);
    float l[6];
    #pragma unroll
    for (int j = 0; j < 6; ++j) l[j] = fc_b[j];
    for (int k = 0; k < F; ++k) {
        const float p = pooled[(size_t)g * F + k] * inv;
        #pragma unroll
        for (int j = 0; j < 6; ++j) l[j] += p * fc_w[k * 6 + j];
    }
    float m = l[0];
    #pragma unroll
    for (int j = 1; j < 6; ++j) m = fmaxf(m, l[j]);
    float s = 0.0f;
    #pragma unroll
    for (int j = 0; j < 6; ++j) s += expf(l[j] - m);
    const float ls = logf(s);
    #pragma unroll
    for (int j = 0; j < 6; ++j) out[(size_t)g * 6 + j] = l[j] - m - ls;
}

// ------------------------------------------------------------------- driver
extern "C" void kernel_launch(void* const* d_in, const int* in_sizes, int n_in,
                              void* d_out, int out_size, void* d_ws, size_t ws_size,
                              hipStream_t stream) {
    const float* x      = (const float*)d_in[0];
    const float* pseudo = (const float*)d_in[1];
    const int*   eidx   = (const int*)  d_in[2];
    const int*   batch  = (const int*)  d_in[3];
    const float* W[4]    = { (const float*)d_in[4],  (const float*)d_in[7],
                             (const float*)d_in[10], (const float*)d_in[13] };
    const float* root[4] = { (const float*)d_in[5],  (const float*)d_in[8],
                             (const float*)d_in[11], (const float*)d_in[14] };
    const float* bias[4] = { (const float*)d_in[6],  (const float*)d_in[9],
                             (const float*)d_in[12], (const float*)d_in[15] };
    const float* fc_w = (const float*)d_in[16];
    const float* fc_b = (const float*)d_in[17];
    float* out = (float*)d_out;

    const int* src = eidx;
    const int* dst = eidx + N_EDGES;

    // workspace layout (floats)
    float* base   = (float*)d_ws;
    const size_t NF = (size_t)N_NODES * F;
    float* y0     = base;
    float* y1     = y0 + NF;
    float* yr     = y1 + NF;
    float* agg    = yr + NF;
    float* h      = agg + NF;
    float* deg    = h + NF;
    float* pooled = deg + N_NODES;
    float* cnt    = pooled + (size_t)N_GRAPHS * F;

    const int TB = 256;
    const int nElem       = N_NODES * F;               // 5,120,000
    const int elemBlocks  = (nElem + TB - 1) / TB;     // 20000
    const int edgeBlocksT = (N_EDGES + TB - 1) / TB;   // 5000 (thread-per-edge)
    const int edgeBlocksW = N_EDGES / 8;               // 160000 (wave-per-edge)
    const int mmBlocks    = (N_NODES / 16) / 4;        // 1250 (4 waves/block)

    // init accumulators every call (ws is poisoned before timing)
    zero_f32<<<elemBlocks, TB, 0, stream>>>(agg, nElem);
    zero_f32<<<elemBlocks, TB, 0, stream>>>(deg, N_NODES);
    zero_f32<<<elemBlocks, TB, 0, stream>>>(pooled, N_GRAPHS * F);
    zero_f32<<<elemBlocks, TB, 0, stream>>>(cnt, N_GRAPHS);
    deg_kernel<<<edgeBlocksT, TB, 0, stream>>>(dst, deg);

    for (int layer = 0; layer < 4; ++layer) {
        if (layer == 0) {
            node_mm3<<<mmBlocks, 128, 0, stream>>>(
                x, W[0], W[0] + 3 * F, root[0], bias[0], y0, y1, yr);
        } else {
            node_mm64<<<mmBlocks, 128, 0, stream>>>(
                h, W[layer], W[layer] + F * F, root[layer], bias[layer],
                y0, y1, yr);
        }
        edge_kernel<<<edgeBlocksW, TB, 0, stream>>>(src, dst, pseudo, y0, y1, agg);
        combine_kernel<<<elemBlocks, TB, 0, stream>>>(agg, yr, deg, h);
    }

    pool_kernel<<<elemBlocks, TB, 0, stream>>>(h, batch, pooled, cnt);
    head_kernel<<<(N_GRAPHS + TB - 1) / TB, TB, 0, stream>>>(pooled, cnt, fc_w, fc_b, out);
}